// MDN_Block_38749194944927
// MI455X (gfx1250) — compile-verified
//
#include <hip/hip_runtime.h>
#include <hip/hip_bf16.h>
#include <math.h>

// ---- problem constants (match reference) ----
#define BB   8
#define NL   64
#define NT   512
#define HH   128
#define NGAU 10
#define EE   2048
#define MM   4096

// ---- output layout (floats, concatenated in reference return order) ----
static constexpr int O_PI = 0;                       // (262144,10)
static constexpr int O_SG = 2621440;                 // (262144,10)
static constexpr int O_MU = 5242880;                 // (262144,10)
static constexpr int O_CB = 7864320;                 // (262144,)
static constexpr int O_AT = 8126464;                 // (512,18)
static constexpr int O_BT = 8135680;                 // (2048,5)
static constexpr int O_CD = 8145920;                 // (4096,)
static constexpr int O_CM = 8150016;                 // (262144,)

// ---- workspace layout (float offsets) ----
static constexpr int W_LG  = 0;        // 512x128
static constexpr int W_PG  = 65536;    // 4096x128
static constexpr int W_LS  = 589824;   // 512x128
static constexpr int W_PS  = 655360;   // 4096x128
static constexpr int W_BT0 = 1179648;  // 512x5
static constexpr int W_BT1 = 1182208;  // 512x5
static constexpr int W_WH  = 1184768;  // 128x32 packed head weights
static constexpr int W_HB  = 1188864;  // 32 packed head bias
static constexpr int W_SS  = 1188896;  // 128 selector bn scale
static constexpr int W_SD  = 1189024;  // 128 selector bn shift

typedef __attribute__((ext_vector_type(2))) float v2f;
typedef __attribute__((ext_vector_type(8))) float v8f;

__device__ __forceinline__ float elu1(float x) {
    return x > 0.f ? x : (expf(x) - 1.f);
}

// -------------------------------------------------------------------------
// Setup: pack pi/sg/mu weights into 128x32 (cols 0-9 pi, 10-19 sg, 20-29 mu,
// 30-31 zero pad), pack biases, precompute selector BN scale/shift.
// -------------------------------------------------------------------------
__global__ void setup_pack(const float* __restrict__ pi_w, const float* __restrict__ sg_w,
                           const float* __restrict__ mu_w, const float* __restrict__ pi_b,
                           const float* __restrict__ sg_b, const float* __restrict__ mu_b,
                           const float* __restrict__ sel_b1, const float* __restrict__ sel_g,
                           const float* __restrict__ sel_be, const float* __restrict__ sel_m,
                           const float* __restrict__ sel_v,
                           float* __restrict__ Whead, float* __restrict__ hBias,
                           float* __restrict__ selS, float* __restrict__ selD) {
    int idx = blockIdx.x * blockDim.x + threadIdx.x;
    if (idx < 128 * 32) {
        int k = idx >> 5, c = idx & 31;
        float v = 0.f;
        if (c < 10)      v = pi_w[k * 10 + c];
        else if (c < 20) v = sg_w[k * 10 + (c - 10)];
        else if (c < 30) v = mu_w[k * 10 + (c - 20)];
        Whead[k * 32 + c] = v;
    } else if (idx < 4096 + 32) {
        int c = idx - 4096;
        float v = 0.f;
        if (c < 10)      v = pi_b[c];
        else if (c < 20) v = sg_b[c - 10];
        else if (c < 30) v = mu_b[c - 20];
        hBias[c] = v;
    } else if (idx < 4128 + 128) {
        int k = idx - 4128;
        float s = sel_g[k] * rsqrtf(sel_v[k] + 1e-5f);
        selS[k] = s;
        selD[k] = (sel_b1[k] - sel_m[k]) * s + sel_be[k];
    }
}

// -------------------------------------------------------------------------
// Generic C(MxN) = A(Mx128) @ W(128xN) [+ bias] using V_WMMA_F32_16X16X4_F32.
// One wave per 16x16 output tile. M must be a multiple of 16; N guarded via
// column clamp + value select (no EXEC manipulation -> WMMA-legal, branchless).
// -------------------------------------------------------------------------
__global__ void __launch_bounds__(32) gemm_wmma_f32(const float* __restrict__ A,
                                                    const float* __restrict__ W,
                                                    const float* __restrict__ bias,
                                                    float* __restrict__ C, int N) {
    const int lane = threadIdx.x & 31;
    const int c = lane & 15, h = lane >> 4;
    const int m0 = blockIdx.x * 16, n0 = blockIdx.y * 16;
    const int col = n0 + c;
    const bool ok = col < N;
    const int colc = ok ? col : (N - 1);   // clamped, always in-bounds

    const float* __restrict__ ap  = A + (m0 + c) * 128 + 2 * h;  // A[m0+c][2h + 4kk]
    const float* __restrict__ wp0 = W + 2 * h * N + colc;        // W[2h + 4kk][colc]
    const float* __restrict__ wp1 = wp0 + N;
    const int wstep = 4 * N;

    v8f acc = {};
#pragma unroll 8
    for (int kk = 0; kk < 32; ++kk) {
        v2f a; a[0] = ap[0]; a[1] = ap[1];
        float w0 = wp0[0];
        float w1 = wp1[0];
        v2f b; b[0] = ok ? w0 : 0.f; b[1] = ok ? w1 : 0.f;
        acc = __builtin_amdgcn_wmma_f32_16x16x4_f32(false, a, false, b, (short)0, acc,
                                                    false, false);
        ap += 4; wp0 += wstep; wp1 += wstep;
    }
    if (ok) {
        float bv = bias ? bias[col] : 0.f;
#pragma unroll
        for (int i = 0; i < 8; ++i) {
            int m = m0 + i + 8 * h;
            C[m * N + col] = acc[i] + bv;
        }
    }
}

// -------------------------------------------------------------------------
// Fused pair kernel: block = 128 threads (4 waves). Block handles fixed
// (b, i) and 64 t's; each wave handles 16 t's.
//   Ch = elu(bn(Lg[li] + Pg[ti]))           (16x128 tile in LDS)
//   heads = Ch @ Whead(128x32) + bias       (2 WMMA f32 chains, 64 WMMAs)
//   epilogue: softmax(pi), elu(sg)+1.1, elu(mu)+1.0, C_batch, C_mask
// Head weights are staged TRANSPOSED in LDS (sWt[col][k]) so each B fragment
// is a contiguous K-pair -> single ds_load_b64 per fragment.
// -------------------------------------------------------------------------
__global__ void __launch_bounds__(128) pair_kernel(const float* __restrict__ Lg,
                                                   const float* __restrict__ Pg,
                                                   const float* __restrict__ Whead,
                                                   const float* __restrict__ hBias,
                                                   const float* __restrict__ mlp_b,
                                                   const float* __restrict__ mlp_g,
                                                   const float* __restrict__ mlp_be,
                                                   const float* __restrict__ mlp_m,
                                                   const float* __restrict__ mlp_v,
                                                   float* __restrict__ out) {
    __shared__ float sWt[32][132];       // transposed head weights, padded
    __shared__ float sBias[32];
    __shared__ float sScale[128];
    __shared__ float sShift[128];
    __shared__ float sLg[128];
    __shared__ float sCh[4][16][132];    // Ch tiles, padded rows
    __shared__ float sStage[4][16][33];  // head outputs staging

    const int tid = threadIdx.x;
    const int bid = blockIdx.x;          // 0..4095
    const int tchunk = bid & 7;          // which group of 64 t's
    const int li = bid >> 3;             // 0..511  (b*64 + i)
    const int b = li >> 6;

    for (int i = tid; i < 128 * 32; i += 128) sWt[i & 31][i >> 5] = Whead[i];
    if (tid < 32) sBias[tid] = hBias[tid];
    if (tid < 128) {
        float s = mlp_g[tid] * rsqrtf(mlp_v[tid] + 1e-5f);
        sScale[tid] = s;
        sShift[tid] = (mlp_b[tid] - mlp_m[tid]) * s + mlp_be[tid];
        sLg[tid] = Lg[li * HH + tid];
    }
    __syncthreads();

    const int w = tid >> 5, lane = tid & 31;
    const int t0 = tchunk * 64 + w * 16;
    const float* __restrict__ PgBase = Pg + (b * NT + t0) * HH;

    // Build Ch tile: 16 rows x 128 cols, coalesced Pg reads.
#pragma unroll 4
    for (int k = 0; k < 64; ++k) {
        int idx = lane + 32 * k;
        int row = idx >> 7, colk = idx & 127;
        float x = sLg[colk] + PgBase[row * HH + colk];
        x = x * sScale[colk] + sShift[colk];
        sCh[w][row][colk] = elu1(x);
    }
    __syncthreads();

    const int c = lane & 15, h = lane >> 4;
    v8f acc0 = {}, acc1 = {};
#pragma unroll 4
    for (int kk = 0; kk < 32; ++kk) {
        int k0 = kk * 4 + h * 2;
        v2f a;  a[0]  = sCh[w][c][k0];      a[1]  = sCh[w][c][k0 + 1];
        v2f b0; b0[0] = sWt[c][k0];         b0[1] = sWt[c][k0 + 1];
        v2f b1; b1[0] = sWt[16 + c][k0];    b1[1] = sWt[16 + c][k0 + 1];
        acc0 = __builtin_amdgcn_wmma_f32_16x16x4_f32(false, a, false, b0, (short)0, acc0,
                                                     false, false);
        acc1 = __builtin_amdgcn_wmma_f32_16x16x4_f32(false, a, false, b1, (short)0, acc1,
                                                     false, false);
    }
#pragma unroll
    for (int i = 0; i < 8; ++i) {
        int m = i + 8 * h;
        sStage[w][m][c]      = acc0[i] + sBias[c];
        sStage[w][m][16 + c] = acc1[i] + sBias[16 + c];
    }
    __syncthreads();

    const int rbase = li * NT + t0;
    if (lane < 16) {
        int row = lane;
        int r = rbase + row;
        float logits[10];
        float mx = -1e30f;
#pragma unroll
        for (int j = 0; j < 10; ++j) { logits[j] = sStage[w][row][j]; mx = fmaxf(mx, logits[j]); }
        float sum = 0.f;
#pragma unroll
        for (int j = 0; j < 10; ++j) { logits[j] = expf(logits[j] - mx); sum += logits[j]; }
        float inv = 1.f / sum;
#pragma unroll
        for (int j = 0; j < 10; ++j) out[O_PI + r * 10 + j] = logits[j] * inv;
        out[O_CB + r] = (float)b;
        out[O_CM + r] = 1.0f;
    } else {
        int row = lane - 16;
        int r = rbase + row;
#pragma unroll
        for (int j = 0; j < 10; ++j)
            out[O_SG + r * 10 + j] = elu1(sStage[w][row][10 + j]) + 1.1f;
#pragma unroll
        for (int j = 0; j < 10; ++j)
            out[O_MU + r * 10 + j] = elu1(sStage[w][row][20 + j]) + 1.0f;
    }
}

// -------------------------------------------------------------------------
// Donor selector: one wave per donor (4096 donors).
// sel_h = elu(bn(Ls[li]+Ps[ti])); cd = sigmoid(sel_h . sel_w2 + sel_b2)
// -------------------------------------------------------------------------
__global__ void __launch_bounds__(256) selector_kernel(const int* __restrict__ donar_idx,
                                                       const float* __restrict__ Ls,
                                                       const float* __restrict__ Ps,
                                                       const float* __restrict__ selS,
                                                       const float* __restrict__ selD,
                                                       const float* __restrict__ sel_w2,
                                                       const float* __restrict__ sel_b2,
                                                       float* __restrict__ out) {
    int w = threadIdx.x >> 5, lane = threadIdx.x & 31;
    int m = blockIdx.x * 8 + w;
    int r = donar_idx[m];
    int b = r / (NL * NT);
    int rem = r - b * (NL * NT);
    int i = rem >> 9;        // / NT
    int t = rem & (NT - 1);
    const float* __restrict__ lrow = Ls + (b * NL + i) * HH;
    const float* __restrict__ prow = Ps + (b * NT + t) * HH;
    float partial = 0.f;
#pragma unroll
    for (int e = 0; e < 4; ++e) {
        int col = lane + 32 * e;
        float x = (lrow[col] + prow[col]) * selS[col] + selD[col];
        partial += elu1(x) * sel_w2[col];
    }
#pragma unroll
    for (int off = 16; off >= 1; off >>= 1) partial += __shfl_xor(partial, off, 32);
    if (lane == 0) {
        float z = partial + sel_b2[0];
        out[O_CD + m] = 1.f / (1.f + expf(-z));
    }
}

// -------------------------------------------------------------------------
// bond_types[e,j] = Bt0[ei0[e],j] + Bt1[ei1[e],j] + bt_b[j]
// -------------------------------------------------------------------------
__global__ void bond_kernel(const int* __restrict__ edge_index,
                            const float* __restrict__ Bt0, const float* __restrict__ Bt1,
                            const float* __restrict__ bt_b, float* __restrict__ out) {
    int idx = blockIdx.x * blockDim.x + threadIdx.x;
    if (idx >= EE * 5) return;
    int e = idx / 5, j = idx - e * 5;
    int s0 = edge_index[e];
    int s1 = edge_index[EE + e];
    out[O_BT + idx] = Bt0[s0 * 5 + j] + Bt1[s1 * 5 + j] + bt_b[j];
}

// -------------------------------------------------------------------------
extern "C" void kernel_launch(void* const* d_in, const int* in_sizes, int n_in,
                              void* d_out, int out_size, void* d_ws, size_t ws_size,
                              hipStream_t stream) {
    (void)in_sizes; (void)n_in; (void)out_size; (void)ws_size;

    const float* lig_s  = (const float*)d_in[0];
    const float* pro_s  = (const float*)d_in[1];
    const int*   donar  = (const int*)d_in[4];
    const int*   eidx   = (const int*)d_in[5];
    const float* mlp_w  = (const float*)d_in[6];
    const float* mlp_b  = (const float*)d_in[7];
    const float* mlp_g  = (const float*)d_in[8];
    const float* mlp_be = (const float*)d_in[9];
    const float* mlp_m  = (const float*)d_in[10];
    const float* mlp_v  = (const float*)d_in[11];
    const float* sel_w1 = (const float*)d_in[12];
    const float* sel_b1 = (const float*)d_in[13];
    const float* sel_g  = (const float*)d_in[14];
    const float* sel_be = (const float*)d_in[15];
    const float* sel_m  = (const float*)d_in[16];
    const float* sel_v  = (const float*)d_in[17];
    const float* sel_w2 = (const float*)d_in[18];
    const float* sel_b2 = (const float*)d_in[19];
    const float* pi_w   = (const float*)d_in[20];
    const float* pi_b   = (const float*)d_in[21];
    const float* sg_w   = (const float*)d_in[22];
    const float* sg_b   = (const float*)d_in[23];
    const float* mu_w   = (const float*)d_in[24];
    const float* mu_b   = (const float*)d_in[25];
    const float* at_w   = (const float*)d_in[26];
    const float* at_b   = (const float*)d_in[27];
    const float* bt_w   = (const float*)d_in[28];
    const float* bt_b   = (const float*)d_in[29];

    float* out = (float*)d_out;
    float* ws  = (float*)d_ws;
    float* Lg  = ws + W_LG;
    float* Pg  = ws + W_PG;
    float* Ls  = ws + W_LS;
    float* Ps  = ws + W_PS;
    float* Bt0 = ws + W_BT0;
    float* Bt1 = ws + W_BT1;
    float* Wh  = ws + W_WH;
    float* hB  = ws + W_HB;
    float* sS  = ws + W_SS;
    float* sD  = ws + W_SD;

    setup_pack<<<17, 256, 0, stream>>>(pi_w, sg_w, mu_w, pi_b, sg_b, mu_b,
                                       sel_b1, sel_g, sel_be, sel_m, sel_v,
                                       Wh, hB, sS, sD);

    // Stage 1: low-rank decomposition GEMMs (all K=128, WMMA f32 16x16x4)
    gemm_wmma_f32<<<dim3(32, 8),  32, 0, stream>>>(lig_s, mlp_w,            nullptr, Lg, HH);
    gemm_wmma_f32<<<dim3(256, 8), 32, 0, stream>>>(pro_s, mlp_w + HH * HH,  nullptr, Pg, HH);
    gemm_wmma_f32<<<dim3(32, 8),  32, 0, stream>>>(lig_s, sel_w1,           nullptr, Ls, HH);
    gemm_wmma_f32<<<dim3(256, 8), 32, 0, stream>>>(pro_s, sel_w1 + HH * HH, nullptr, Ps, HH);
    gemm_wmma_f32<<<dim3(32, 2),  32, 0, stream>>>(lig_s, at_w, at_b, out + O_AT, 18);
    gemm_wmma_f32<<<dim3(32, 1),  32, 0, stream>>>(lig_s, bt_w,          nullptr, Bt0, 5);
    gemm_wmma_f32<<<dim3(32, 1),  32, 0, stream>>>(lig_s, bt_w + HH * 5, nullptr, Bt1, 5);

    // Stage 2: fused pair MDN (pi/sigma/mu/C_batch/C_mask)
    pair_kernel<<<4096, 128, 0, stream>>>(Lg, Pg, Wh, hB, mlp_b, mlp_g, mlp_be, mlp_m,
                                          mlp_v, out);

    // Stage 3: donor selector + bond types
    selector_kernel<<<512, 256, 0, stream>>>(donar, Ls, Ps, sS, sD, sel_w2, sel_b2, out);
    bond_kernel<<<40, 256, 0, stream>>>(eidx, Bt0, Bt1, bt_b, out);
}